// HardSort_55774445306518
// MI455X (gfx1250) — compile-verified
//
#include <hip/hip_runtime.h>

// HardSort on MI455X (gfx1250, wave32).
//
// out[b,r,c] = max(h_r - |scores[b,c] - sorted[b,r]|, 0) / h_r
//   h_r = 0.5 * min(sorted[r-1]-sorted[r], sorted[r]-sorted[r+1])   (edges one-sided)
//
// Store-bound (256 MB out @ 23.3 TB/s ~= 11 us). Strategy:
//  - per block: stage row in LDS, bitonic sort descending in LDS
//  - diff tiles via V_WMMA_F32_16X16X4_F32 (rank-2 product [scores|1] x [[1],[-sorted]])
//    with matrix-col == output row so each lane owns one output row (h_r scalar/lane)
//  - epilogue (abs/max/mul) on VALU, co-executes with matrix pipe
//  - tile loop unrolled x4: 4 independent ds_load -> 4 independent wmma, no
//    per-tile EXEC masking (unconditional LDS broadcast read + cndmask)
//  - two non-temporal global_store_b128 per lane per tile (streaming output)

typedef float v2f __attribute__((ext_vector_type(2)));
typedef float v4f __attribute__((ext_vector_type(4)));
typedef float v8f __attribute__((ext_vector_type(8)));

#define BATCH 64
#define N     1024

__global__ __launch_bounds__(256) void hardsort_wmma_kernel(
    const float* __restrict__ scores, float* __restrict__ out) {
  __shared__ float s_scores[N];   // original order (columns)
  __shared__ float s_sorted[N];   // descending sort (rows)

  const int tid = threadIdx.x;
  const int b   = blockIdx.x >> 4;   // batch index
  const int rg  = blockIdx.x & 15;   // row-group: rows [rg*64, rg*64+64)

  // ---- stage the row into LDS ----
  const float* row = scores + (size_t)b * N;
  for (int i = tid; i < N; i += 256) {
    float v = row[i];
    s_scores[i] = v;
    s_sorted[i] = v;
  }
  __syncthreads();

  // ---- bitonic sort, descending, 1024 elems, 512 pairs/stage over 256 threads ----
  for (int k = 2; k <= N; k <<= 1) {
    for (int j = k >> 1; j > 0; j >>= 1) {
      for (int t = tid; t < N / 2; t += 256) {
        int i = ((t & ~(j - 1)) << 1) | (t & (j - 1));
        int p = i | j;
        float a = s_sorted[i];
        float c = s_sorted[p];
        bool descSeg = (i & k) == 0;
        bool doSwap  = descSeg ? (a < c) : (a > c);
        if (doSwap) { s_sorted[i] = c; s_sorted[p] = a; }
      }
      __syncthreads();
    }
  }

  // ---- per-wave tile assignment ----
  const int lane = tid & 31;
  const int w    = tid >> 5;        // 8 waves
  const int rt   = w & 3;           // row tile 0..3  (16 rows each)
  const int ch   = w >> 2;          // column half 0..1 (512 cols each)
  const int r0   = rg * 64 + rt * 16;
  const int nn   = lane & 15;       // D-matrix column index == local output row
  const int r    = r0 + nn;
  const bool lo  = (lane < 16);     // lanes that carry K=0/1 of A and B

  // per-lane row constants (loaded once, loop-invariant)
  const float s_r   = s_sorted[r];
  const float gprev = (r > 0)     ? (s_sorted[r - 1] - s_r) : __builtin_inff();
  const float gnext = (r < N - 1) ? (s_r - s_sorted[r + 1]) : __builtin_inff();
  const float h     = 0.5f * fminf(gprev, gnext);
  const float inv   = 1.0f / h;

  // B (4x16 f32): lanes 0-15 hold K=0 (VGPR0) / K=1 (VGPR1); lanes 16-31 hold K=2,3 = 0
  //   B[0,n] = 1, B[1,n] = -sorted[r0+n]
  v2f bm;
  bm.x = lo ? 1.0f : 0.0f;
  bm.y = lo ? -s_r : 0.0f;
  const float a_hi = lo ? 1.0f : 0.0f;   // A[m,1] = 1 lives in lanes 0-15 only

  // D layout: lane<16 -> (n=lane,  m=v)    ; lane>=16 -> (n=lane-16, m=v+8)
  // output col = c0 + m  ->  8 consecutive floats per lane
  float* outrow = out + ((size_t)b * N + r) * N + (lo ? 0 : 8);

  const int cbeg = ch * 512;
#pragma unroll 4
  for (int c0 = cbeg; c0 < cbeg + 512; c0 += 16) {
    // A (16x4 f32): lanes 0-15 hold K=0/1; lanes 16-31 hold K=2/3 = 0.
    // Unconditional LDS read: lanes 16-31 hit the same words as lanes 0-15
    // (broadcast, conflict-free), then mask with cndmask -- no EXEC juggling.
    float av = s_scores[c0 + nn];
    v2f am;
    am.x = lo ? av : 0.0f;
    am.y = a_hi;

    v8f acc = {};
    // D[m,n] = scores[c0+m]*1 + 1*(-sorted[r0+n]) ; exact fp32 subtraction on matrix pipe
    acc = __builtin_amdgcn_wmma_f32_16x16x4_f32(
        /*neg_a=*/false, am, /*neg_b=*/false, bm,
        /*c_mod=*/(short)0, acc, /*reuse_a=*/false, /*reuse_b=*/false);

    v4f o0, o1;
#pragma unroll
    for (int v = 0; v < 4; ++v)
      o0[v] = fmaxf(h - __builtin_fabsf(acc[v]), 0.0f) * inv;
#pragma unroll
    for (int v = 0; v < 4; ++v)
      o1[v] = fmaxf(h - __builtin_fabsf(acc[v + 4]), 0.0f) * inv;

    // streaming output: bypass-cache hint, 2x global_store_b128 per lane
    __builtin_nontemporal_store(o0, (v4f*)(outrow + c0));
    __builtin_nontemporal_store(o1, (v4f*)(outrow + c0 + 4));
  }
}

extern "C" void kernel_launch(void* const* d_in, const int* in_sizes, int n_in,
                              void* d_out, int out_size, void* d_ws, size_t ws_size,
                              hipStream_t stream) {
  (void)in_sizes; (void)n_in; (void)d_ws; (void)ws_size; (void)out_size;
  const float* scores = (const float*)d_in[0];
  float* out = (float*)d_out;
  hipLaunchKernelGGL(hardsort_wmma_kernel, dim3(BATCH * 16), dim3(256), 0, stream,
                     scores, out);
}